// SepKanModel_78039555769134
// MI455X (gfx1250) — compile-verified
//
#include <hip/hip_runtime.h>
#include <hip/hip_bf16.h>

// ---------------- problem constants (from reference) ----------------
#define NBATCH   8
#define NFEAT    1000
#define NPARAM   4576          // KAN_PARAMS_NUM
#define HW       50176         // 224*224
#define KB       11            // GRID_SIZE + SPLINE_ORDER
#define RPI      12            // 11 basis + 1 silu per input dim
// merged weight matrix sizes per batch (rows x 16 cols, layer2 padded to 16)
#define WM0_ROWS (3*RPI)       // 36
#define WM1_ROWS (16*RPI)      // 192
#define WM2_ROWS (16*RPI)      // 192
#define WM_PER_B (16*(WM0_ROWS + WM1_ROWS + WM2_ROWS))   // 6720 floats

// ---------------- main-kernel tiling ----------------
#define TOK   64               // tokens per block (one 16-token WMMA tile per wave)
#define SY    17               // y ping-pong LDS stride (bank-conflict pad)
#define FSTR  198              // feature LDS stride (step 6 mod 64 -> conflict-free)

// ---------------- FC (WMMA) kernel tiling ----------------
#define FCCOLS 64              // output columns per block (16 per wave)
#define FCKT   100             // K tile (1000 = 10 tiles, multiple of 4)
#define FSTRF  1002            // features LDS stride (step 42 mod 64 -> conflict-free)
#define WTSTR  101             // fc_w tile LDS stride (step 37 mod 64 -> conflict-free)

typedef __attribute__((ext_vector_type(2))) float v2f;
typedef __attribute__((ext_vector_type(8))) float v8f;

// -------------------------------------------------------------------
// Kernel 1 (WMMA GEMM): w[b][j] = sum_k features[b][k]*fc_w[j][k] + fc_b[j]
//   M = 16 (8 batches + 8 zero rows), N = 64 cols/block, K = 1000.
//   fc_w staged through LDS in coalesced runs; f32 16x16x4 WMMA.
// -------------------------------------------------------------------
__global__ void kan_fc_wmma_kernel(const float* __restrict__ features,
                                   const float* __restrict__ fc_w,
                                   const float* __restrict__ fc_b,
                                   float* __restrict__ w_out) {
    extern __shared__ float smem[];
    float* fs = smem;                  // [16][FSTRF]  padded batch-major features
    float* wt = fs + 16 * FSTRF;       // [FCCOLS][WTSTR] fc_w K-tile (B matrix, col-major rows)

    const int tid  = threadIdx.x;
    const int lane = tid & 31;
    const int wv   = tid >> 5;
    const int m    = lane & 15;
    const int hi   = lane >> 4;
    const int j0   = blockIdx.x * FCCOLS;

    // stage features, zero-pad batches 8..15
    for (int idx = tid; idx < 16 * NFEAT; idx += 128) {
        const int mm = idx / NFEAT;
        const int k  = idx % NFEAT;
        fs[mm * FSTRF + k] = (mm < NBATCH) ? features[mm * NFEAT + k] : 0.0f;
    }

    const int  jc   = j0 + wv * 16 + m;                 // this lane's output column
    const float bias = (jc < NPARAM) ? fc_b[jc] : 0.0f;
    v8f acc;
#pragma unroll
    for (int r = 0; r < 8; ++r) acc[r] = bias;          // bias is per-column -> same in all rows

    for (int kt = 0; kt < NFEAT; kt += FCKT) {
        __syncthreads();                                // features ready / prev tile consumed
        // coalesced stage of fc_w tile: cols j0..j0+63, K rows kt..kt+FCKT-1
        for (int e = tid; e < FCCOLS * FCKT; e += 128) {
            const int c = e / FCKT;
            const int k = e % FCKT;
            const int j = j0 + c;
            wt[c * WTSTR + k] = (j < NPARAM) ? fc_w[(size_t)j * NFEAT + kt + k] : 0.0f;
        }
        __syncthreads();
        const float* arow = &fs[m * FSTRF + kt];        // A row = batch m
        const float* brow = &wt[(wv * 16 + m) * WTSTR]; // B col = lane's column
        for (int k0 = 0; k0 < FCKT; k0 += 4) {
            const int ka = k0 + 2 * hi;                 // lanes 0-15: K=k0,k0+1; 16-31: +2
            v2f a, bb;
            a.x  = arow[ka];
            a.y  = arow[ka + 1];
            bb.x = brow[ka];
            bb.y = brow[ka + 1];
            acc = __builtin_amdgcn_wmma_f32_16x16x4_f32(
                    false, a, false, bb, (short)0, acc, false, false);
        }
    }
    // C/D: VGPR r -> row r (lanes 0-15) / r+8 (lanes 16-31); batches live in rows 0-7
    if (hi == 0 && jc < NPARAM) {
#pragma unroll
        for (int r = 0; r < 8; ++r)
            w_out[r * NPARAM + jc] = acc[r];
    }
}

// -------------------------------------------------------------------
// Kernel 2: build merged per-batch weight matrices
//   row r = i*12+k, col o:  k<11 -> coef[b,i,o,k]*uw[b,i,o], k==11 -> rw[b,i,o]
//   layer2 cols o>=3 are zero-padded.
// -------------------------------------------------------------------
__global__ void kan_wm_kernel(const float* __restrict__ w,
                              float* __restrict__ wm) {
    const int e = blockIdx.x * blockDim.x + threadIdx.x;
    const int total = NBATCH * WM_PER_B;
    if (e >= total) return;
    const int b   = e / WM_PER_B;
    const int rem = e % WM_PER_B;

    int layer, idx;
    if (rem < 16 * WM0_ROWS)                   { layer = 0; idx = rem; }
    else if (rem < 16 * (WM0_ROWS + WM1_ROWS)) { layer = 1; idx = rem - 16 * WM0_ROWS; }
    else                                       { layer = 2; idx = rem - 16 * (WM0_ROWS + WM1_ROWS); }

    const int DO_[3] = {16, 16, 3};
    const int I0[3]  = {0,   624, 3952};   // coef offsets (PARAM_IDX)
    const int I1[3]  = {528, 3440, 4480};  // uw offsets
    const int I2[3]  = {576, 3696, 4528};  // rw offsets

    const int r = idx >> 4;          // row
    const int o = idx & 15;          // col
    const int i = r / RPI;
    const int k = r % RPI;

    float val = 0.0f;
    const int do_ = DO_[layer];
    if (o < do_) {
        const float* wb = w + (size_t)b * NPARAM;
        const int io = i * do_ + o;
        if (k < KB) val = wb[I0[layer] + io * KB + k] * wb[I1[layer] + io];
        else        val = wb[I2[layer] + io];
    }
    wm[e] = val;
}

// -------------------------------------------------------------------
// cubic B-spline basis (uniform grid, h=0.25, knots -1.75..1.75) + silu
// f[0..10] = basis, f[11] = silu(x)
// -------------------------------------------------------------------
__device__ __forceinline__ void bspline12(float x, float* f) {
    float b[14];
#pragma unroll
    for (int j = 0; j < 14; ++j) {
        const float g0 = -1.75f + 0.25f * j;
        b[j] = (x >= g0 && x < g0 + 0.25f) ? 1.0f : 0.0f;
    }
#pragma unroll
    for (int p = 1; p <= 3; ++p) {
        const float inv = 1.0f / (0.25f * p);   // uniform-grid denominator
#pragma unroll 13
        for (int j = 0; j < 14 - p; ++j) {
            const float gj = -1.75f + 0.25f * j;
            const float left  = (x - gj) * inv * b[j];
            const float right = ((gj + 0.25f * (p + 1)) - x) * inv * b[j + 1];
            b[j] = left + right;
        }
    }
#pragma unroll
    for (int k = 0; k < KB; ++k) f[k] = b[k];
    f[KB] = x / (1.0f + __expf(-x));            // silu
}

// -------------------------------------------------------------------
// One KAN layer for 64 tokens: feature build (LDS) + WMMA f32 16x16x4.
// Single pass; every wave owns one 16-token tile (full WGP utilization).
// -------------------------------------------------------------------
template <int DI>
__device__ __forceinline__ void kan_layer(const float* __restrict__ wmL,  // [R][16] LDS
                                          const float* yin,               // [TOK][SY] LDS
                                          float* yout,                    // [TOK][SY] LDS
                                          float* feat,                    // [TOK][FSTR] LDS
                                          int tid) {
    const int R    = DI * RPI;
    const int lane = tid & 31;
    const int wv   = tid >> 5;
    const int m    = lane & 15;
    const int hi   = lane >> 4;

    // ---- feature build: one (token, in-dim) pair per task ----
    for (int task = tid; task < TOK * DI; task += 128) {
        const int t = task & (TOK - 1);
        const int i = task >> 6;
        float f[RPI];
        bspline12(yin[t * SY + i], f);
        float* dst = &feat[t * FSTR + i * RPI];
#pragma unroll
        for (int k = 0; k < RPI; ++k) dst[k] = f[k];
    }
    __syncthreads();

    // ---- WMMA: each wave one 16-token tile, 16 output cols ----
    const int tloc = wv * 16;
    v8f acc = {0.f, 0.f, 0.f, 0.f, 0.f, 0.f, 0.f, 0.f};
    const float* arow = &feat[(tloc + m) * FSTR];
    for (int k0 = 0; k0 < R; k0 += 4) {
        const int ka = k0 + 2 * hi;          // lanes 0-15: K=k0,k0+1; 16-31: K=k0+2,k0+3
        v2f a, bb;
        a.x  = arow[ka];
        a.y  = arow[ka + 1];
        bb.x = wmL[ka * 16 + m];
        bb.y = wmL[(ka + 1) * 16 + m];
        acc = __builtin_amdgcn_wmma_f32_16x16x4_f32(
                false, a, false, bb, (short)0, acc, false, false);
    }
    // C/D layout: VGPR r -> row r (lanes 0-15) / row r+8 (lanes 16-31), col = lane&15
#pragma unroll
    for (int r = 0; r < 8; ++r)
        yout[(tloc + r + 8 * hi) * SY + m] = acc[r];
    __syncthreads();
}

// -------------------------------------------------------------------
// Kernel 3: per-token 3-layer KAN pipeline, 64 tokens / 128 threads per block
// -------------------------------------------------------------------
__global__ void kan_main_kernel(const float* __restrict__ x,
                                const float* __restrict__ wm,
                                float* __restrict__ out) {
    extern __shared__ float smem[];
    float* wms  = smem;                        // WM_PER_B = 6720 floats
    float* yA   = wms + WM_PER_B;              // TOK*SY
    float* yB   = yA + TOK * SY;               // TOK*SY
    float* feat = yB + TOK * SY;               // TOK*FSTR

    const int tid   = threadIdx.x;
    const int b     = blockIdx.x / (HW / TOK);
    const int chunk = blockIdx.x % (HW / TOK);
    const int hw0   = chunk * TOK;

    // stage per-batch merged weights into LDS (L2-resident, 26.9 KB)
    const float* wmb = wm + (size_t)b * WM_PER_B;
    for (int i = tid; i < WM_PER_B; i += 128) wms[i] = wmb[i];

    // stage input channels (x is [B,3,H,W] -> token-major [t][c])
#pragma unroll
    for (int c = 0; c < 3; ++c)
        for (int t = tid; t < TOK; t += 128)
            yA[t * SY + c] = x[((size_t)(b * 3 + c)) * HW + hw0 + t];
    __syncthreads();

    kan_layer<3>(wms,                               yA, yB, feat, tid);  // 3  -> 16
    kan_layer<16>(wms + 16 * WM0_ROWS,              yB, yA, feat, tid);  // 16 -> 16
    kan_layer<16>(wms + 16 * (WM0_ROWS + WM1_ROWS), yA, yB, feat, tid);  // 16 -> 3 (padded)

    // write output [B,3,H,W]
#pragma unroll
    for (int c = 0; c < 3; ++c)
        for (int t = tid; t < TOK; t += 128)
            out[((size_t)(b * 3 + c)) * HW + hw0 + t] = yB[t * SY + c];
}

// -------------------------------------------------------------------
extern "C" void kernel_launch(void* const* d_in, const int* in_sizes, int n_in,
                              void* d_out, int out_size, void* d_ws, size_t ws_size,
                              hipStream_t stream) {
    const float* x        = (const float*)d_in[0];
    const float* features = (const float*)d_in[1];
    const float* fc_w     = (const float*)d_in[2];
    const float* fc_b     = (const float*)d_in[3];
    float* out    = (float*)d_out;
    float* w_buf  = (float*)d_ws;                       // NBATCH*NPARAM floats
    float* wm_buf = w_buf + (size_t)NBATCH * NPARAM;    // NBATCH*WM_PER_B floats

    // FC GEMM: 72 blocks x 128 threads, ~90 KB LDS (gfx1250 WGP has 320 KB)
    const size_t fc_smem = (16 * FSTRF + FCCOLS * WTSTR) * sizeof(float);
    kan_fc_wmma_kernel<<<(NPARAM + FCCOLS - 1) / FCCOLS, 128, fc_smem, stream>>>(
        features, fc_w, fc_b, w_buf);

    kan_wm_kernel<<<(NBATCH * WM_PER_B + 255) / 256, 256, 0, stream>>>(w_buf, wm_buf);

    const size_t smem_bytes = (WM_PER_B + 2 * TOK * SY + TOK * FSTR) * sizeof(float); // 86272 B
    kan_main_kernel<<<NBATCH * (HW / TOK), 128, smem_bytes, stream>>>(x, wm_buf, out);
}